// DynamicKGE_10565619548591
// MI455X (gfx1250) — compile-verified
//
#include <hip/hip_runtime.h>

// RGCN (basis-decomposed) x2 layers for MI455X / gfx1250, wave32.
// Strategy: transform-then-gather. Dense GEMMs via V_WMMA_F32_16X16X4_F32
// (fp32 kept: workload is bandwidth-bound on the edge gather/scatter phase,
// ~1.8 GB/layer @ 23.3 TB/s, so lower precision buys nothing and fp32 matches
// the reference numerics).

typedef __attribute__((ext_vector_type(2))) float v2f;
typedef __attribute__((ext_vector_type(8))) float v8f;
typedef __attribute__((ext_vector_type(4))) float f32x4;

#define DIM 128
#define NB  4

// ---------------------------------------------------------------------------
// GEMM: blockIdx.y = weight index wi (0..3 = basis[wi] -> Y[:, wi*128:+128],
// 4 = root -> Rout). Block = 256 threads = 8 waves; wave w computes rows
// [blockIdx.x*128 + w*16, +16) x all 128 cols. Weight (128x128 f32 = 64 KB)
// staged in LDS once per block.
// ---------------------------------------------------------------------------
__global__ __launch_bounds__(256)
void rgcn_basis_gemm(const float* __restrict__ X, const int* __restrict__ rowidx,
                     const float* __restrict__ basis, const float* __restrict__ root,
                     float* __restrict__ Y, float* __restrict__ Rout, int N)
{
    __shared__ float Wlds[DIM * DIM]; // 64 KB of the 320 KB WGP LDS

    const int wi = blockIdx.y;
    const float* __restrict__ Wg = (wi < NB) ? (basis + (size_t)wi * DIM * DIM) : root;

    const int tid = threadIdx.x;
    {
        const f32x4* __restrict__ s = (const f32x4*)Wg;
        f32x4* d = (f32x4*)Wlds;
        #pragma unroll
        for (int i = 0; i < (DIM * DIM / 4) / 256; ++i)
            d[tid + i * 256] = s[tid + i * 256];   // global_load_b128 -> ds_store_b128
    }
    __syncthreads();

    const int wave = tid >> 5;
    const int lane = tid & 31;
    const int half = lane >> 4;       // 0: K={k,k+1}, 1: K={k+2,k+3}
    const int l16  = lane & 15;
    const int row0 = blockIdx.x * 128 + wave * 16;

    // A-fragment row for this lane (clamped for the ragged tail; stores are
    // guarded later so EXEC stays all-1s through the WMMAs).
    int arow  = row0 + l16;
    int arowc = (arow < N) ? arow : (N - 1);
    int srow  = rowidx ? rowidx[arowc] : arowc;   // layer-1 entity gather
    const float* __restrict__ Arow = X + (size_t)srow * DIM;

    v8f acc[8] = {};   // 8 tiles x 8 VGPRs f32 accumulators

    for (int k0 = 0; k0 < DIM; k0 += 4) {
        const int ka = k0 + 2 * half;
        v2f a;
        a.x = Arow[ka];
        a.y = Arow[ka + 1];
        #pragma unroll
        for (int t = 0; t < 8; ++t) {
            const int col = t * 16 + l16;
            v2f b;
            b.x = Wlds[ka * DIM + col];
            b.y = Wlds[(ka + 1) * DIM + col];
            acc[t] = __builtin_amdgcn_wmma_f32_16x16x4_f32(
                false, a, false, b, (short)0, acc[t], false, false);
        }
    }

    // D layout: lanes 0-15 vgpr r -> M=row0+r; lanes 16-31 -> M=row0+8+r; N=t*16+l16.
    float* __restrict__ Ob;
    size_t ostride, coff;
    if (wi < NB) { Ob = Y;    ostride = (size_t)NB * DIM; coff = (size_t)wi * DIM; }
    else         { Ob = Rout; ostride = DIM;              coff = 0; }

    #pragma unroll
    for (int r = 0; r < 8; ++r) {
        const int m = row0 + half * 8 + r;
        if (m < N) {
            #pragma unroll
            for (int t = 0; t < 8; ++t) {
                const int col = t * 16 + l16;
                Ob[(size_t)m * ostride + coff + col] = acc[t][r];
            }
        }
    }
}

// ---------------------------------------------------------------------------
// Edge phase: msg_j = norm * sum_b att[type,b] * Y[src, b, j]; atomic segment
// sum into agg[dst]. 128 threads per edge (one per output channel), 2 edges
// per 256-thread block. Reads are fully coalesced (Y rows are 2 KB contiguous).
// ---------------------------------------------------------------------------
__global__ __launch_bounds__(256)
void edge_combine(const float* __restrict__ Y, const int* __restrict__ src,
                  const int* __restrict__ dst, const int* __restrict__ etype,
                  const float* __restrict__ enorm, const float* __restrict__ att,
                  float* __restrict__ agg, int E)
{
    const int e = blockIdx.x * 2 + (threadIdx.x >> 7);
    const int j = threadIdx.x & (DIM - 1);
    if (e >= E) return;
    const int   s  = src[e];
    const int   d  = dst[e];
    const int   t  = etype[e];
    const float nr = enorm[e];
    const float* __restrict__ a4 = att + (size_t)t * NB;
    const float c0 = a4[0] * nr, c1 = a4[1] * nr, c2 = a4[2] * nr, c3 = a4[3] * nr;
    const float* __restrict__ yr = Y + (size_t)s * (NB * DIM);
    const float m = c0 * yr[j] + c1 * yr[DIM + j] + c2 * yr[2 * DIM + j] + c3 * yr[3 * DIM + j];
    atomicAdd(&agg[(size_t)d * DIM + j], m);
}

// In-degree (shared by both layers).
__global__ __launch_bounds__(256)
void deg_count(const int* __restrict__ dst, float* __restrict__ deg, int E)
{
    const int e = blockIdx.x * 256 + threadIdx.x;
    if (e < E) atomicAdd(&deg[dst[e]], 1.0f);
}

// out = agg / max(deg,1) + x@root + bias  (+ ReLU for layer 1)
__global__ __launch_bounds__(256)
void finalize(const float* __restrict__ agg, const float* __restrict__ deg,
              const float* __restrict__ rout, const float* __restrict__ bias,
              float* __restrict__ out, int N, int do_relu)
{
    const int i = blockIdx.x * 256 + threadIdx.x;
    if (i >= N * DIM) return;
    const int n = i >> 7;
    const int j = i & (DIM - 1);
    float dg = deg[n];
    dg = dg > 1.0f ? dg : 1.0f;
    float v = agg[i] / dg + rout[i] + bias[j];
    out[i] = do_relu ? (v > 0.0f ? v : 0.0f) : v;
}

__global__ __launch_bounds__(256)
void copy_f32(const float* __restrict__ in, float* __restrict__ out, int n)
{
    const int i = blockIdx.x * 256 + threadIdx.x;
    if (i < n) out[i] = in[i];
}

// ---------------------------------------------------------------------------
extern "C" void kernel_launch(void* const* d_in, const int* in_sizes, int n_in,
                              void* d_out, int out_size, void* d_ws, size_t ws_size,
                              hipStream_t stream)
{
    const int*   entity       = (const int*)  d_in[0];
    const int*   edge_index   = (const int*)  d_in[1];
    const int*   edge_type    = (const int*)  d_in[2];
    const float* edge_norm    = (const float*)d_in[3];
    /* d_in[4] = DAD_rel: unused by the reference */
    const float* entity_table = (const float*)d_in[5];
    const float* relation_emb = (const float*)d_in[6];
    const float* basis1       = (const float*)d_in[7];
    const float* att1         = (const float*)d_in[8];
    const float* root1        = (const float*)d_in[9];
    const float* bias1        = (const float*)d_in[10];
    const float* basis2       = (const float*)d_in[11];
    const float* att2         = (const float*)d_in[12];
    const float* root2        = (const float*)d_in[13];
    const float* bias2        = (const float*)d_in[14];

    const int N = in_sizes[0];          // 100000 entities
    const int E = in_sizes[2];          // 600000 edges
    const int R = in_sizes[6] / DIM;    // 500 relations

    const int* src  = edge_index;       // edge_index[0]
    const int* dstv = edge_index + E;   // edge_index[1]

    // Workspace layout (floats): Y [N*512] | agg [N*128] | rout [N*128] | deg [N]
    float* ws   = (float*)d_ws;
    float* Y    = ws;
    float* agg  = Y    + (size_t)N * NB * DIM;
    float* rout = agg  + (size_t)N * DIM;
    float* deg  = rout + (size_t)N * DIM;

    float* out = (float*)d_out;
    float* h1  = out;  // reuse d_out's node region as layer-1 activation store

    const dim3 ggemm((N + 127) / 128, NB + 1);   // 782 x 5 blocks
    const int  gedge = (E + 1) / 2;
    const int  gfin  = (N * DIM + 255) / 256;

    hipMemsetAsync(deg, 0, sizeof(float) * (size_t)N, stream);
    hipMemsetAsync(agg, 0, sizeof(float) * (size_t)N * DIM, stream);
    deg_count<<<(E + 255) / 256, 256, 0, stream>>>(dstv, deg, E);

    // ---- Layer 1 (input = entity_table[entity], ReLU) ----
    rgcn_basis_gemm<<<ggemm, 256, 0, stream>>>(entity_table, entity, basis1, root1, Y, rout, N);
    edge_combine<<<gedge, 256, 0, stream>>>(Y, src, dstv, edge_type, edge_norm, att1, agg, E);
    finalize<<<gfin, 256, 0, stream>>>(agg, deg, rout, bias1, h1, N, 1);

    // ---- Layer 2 (input = h1, no ReLU), written in place over h1 ----
    hipMemsetAsync(agg, 0, sizeof(float) * (size_t)N * DIM, stream);
    rgcn_basis_gemm<<<ggemm, 256, 0, stream>>>(h1, nullptr, basis2, root2, Y, rout, N);
    edge_combine<<<gedge, 256, 0, stream>>>(Y, src, dstv, edge_type, edge_norm, att2, agg, E);
    finalize<<<gfin, 256, 0, stream>>>(agg, deg, rout, bias2, out, N, 0);

    // ---- Tail output: relation_emb passthrough ----
    copy_f32<<<(R * DIM + 255) / 256, 256, 0, stream>>>(relation_emb, out + (size_t)N * DIM, R * DIM);
}